// HashEncodingLevel_47785806135526
// MI455X (gfx1250) — compile-verified
//
#include <hip/hip_runtime.h>
#include <stdint.h>

#define TABLE_SIZE 524288u
#define TABLE_MASK (TABLE_SIZE - 1u)
#define PRIME_Y 2654435761u
#define PRIME_Z 805459861u
#define BLK 256

#define GLOBAL_AS __attribute__((address_space(1)))
#define LDS_AS    __attribute__((address_space(3)))

// ---------------------------------------------------------------------------
// Stage 3*BLK contiguous coord floats into LDS, coalesced. Uses the gfx1250
// async global->LDS copy path when the toolchain exposes it; falls back to
// plain loads otherwise (same semantics).
// ---------------------------------------------------------------------------
__device__ __forceinline__ void stage_coords(const float* __restrict__ gsrc,
                                             float* sc, int count) {
#pragma unroll
  for (int i = 0; i < 3; ++i) {
    int off = (int)threadIdx.x + i * BLK;
    if (off < count) {
#if __has_builtin(__builtin_amdgcn_global_load_async_to_lds_b32)
      __builtin_amdgcn_global_load_async_to_lds_b32(
          (GLOBAL_AS int*)(gsrc + off),
          (LDS_AS int*)(sc + off),
          /*imm offset*/ 0, /*cpol*/ 0);
#else
      sc[off] = gsrc[off];
#endif
    }
  }
#if __has_builtin(__builtin_amdgcn_global_load_async_to_lds_b32)
#if __has_builtin(__builtin_amdgcn_s_wait_asynccnt)
  __builtin_amdgcn_s_wait_asynccnt(0);
#else
  asm volatile("s_wait_asynccnt 0" ::: "memory");
#endif
#endif
  __syncthreads();
}

// ---------------------------------------------------------------------------
// Pre-pass: interleave grid (2, TABLE_SIZE) SoA -> float2 AoS table in ws.
// Halves the gather count in the main kernel (one B64 gather per corner).
// ---------------------------------------------------------------------------
__global__ void __launch_bounds__(BLK)
interleave_kernel(const float* __restrict__ grid, float2* __restrict__ tab) {
  uint32_t i = blockIdx.x * BLK + threadIdx.x;
  if (i < TABLE_SIZE) {
    tab[i] = make_float2(grid[i], grid[i + TABLE_SIZE]);
  }
}

// ---------------------------------------------------------------------------
// Common per-point math.
// ---------------------------------------------------------------------------
#define DECODE_POINT(cx, cy, cz)                                              \
  float sx = (cx) * 127.0f, sy = (cy) * 127.0f, sz = (cz) * 127.0f;           \
  float fx = floorf(sx), fy = floorf(sy), fz = floorf(sz);                    \
  float wx = sx - fx, wy = sy - fy, wz = sz - fz;                             \
  int x0 = (int)fx; x0 = x0 < 0 ? 0 : (x0 > 126 ? 126 : x0);                  \
  int y0 = (int)fy; y0 = y0 < 0 ? 0 : (y0 > 126 ? 126 : y0);                  \
  int z0 = (int)fz; z0 = z0 < 0 ? 0 : (z0 > 126 ? 126 : z0);                  \
  uint32_t hx0 = (uint32_t)x0, hx1 = (uint32_t)(x0 + 1);                      \
  uint32_t hy0 = (uint32_t)y0 * PRIME_Y, hy1 = (uint32_t)(y0 + 1) * PRIME_Y;  \
  uint32_t hz0 = (uint32_t)z0 * PRIME_Z, hz1 = (uint32_t)(z0 + 1) * PRIME_Z;  \
  float wx0 = 1.0f - wx, wy0 = 1.0f - wy, wz0 = 1.0f - wz;                    \
  uint32_t hxyA = hx0 ^ hy0, hxyB = hx0 ^ hy1;                                \
  uint32_t hxyC = hx1 ^ hy0, hxyD = hx1 ^ hy1;                                \
  float wxyA = wx0 * wy0, wxyB = wx0 * wy;                                    \
  float wxyC = wx * wy0,  wxyD = wx * wy

// ---------------------------------------------------------------------------
// Main kernel, interleaved table in workspace (preferred path).
// ---------------------------------------------------------------------------
__global__ void __launch_bounds__(BLK)
hashenc_interleaved(const float* __restrict__ coords,
                    const float2* __restrict__ tab,
                    float2* __restrict__ out, int n_points) {
  __shared__ float sc[3 * BLK];
  int base = blockIdx.x * BLK;
  int cnt = n_points - base; cnt = cnt > BLK ? BLK : cnt;
  stage_coords(coords + (size_t)base * 3, sc, cnt * 3);

  int n = base + (int)threadIdx.x;
  if (n >= n_points) return;

  float cx = sc[threadIdx.x * 3 + 0];
  float cy = sc[threadIdx.x * 3 + 1];
  float cz = sc[threadIdx.x * 3 + 2];
  DECODE_POINT(cx, cy, cz);

  float a0 = 0.0f, a1 = 0.0f;
#define CORNER(hxy, wxy, hzv, wzv)                                            \
  {                                                                           \
    uint32_t h = ((hxy) ^ (hzv)) & TABLE_MASK;                                \
    float2 g = tab[h];                                                        \
    float w = (wxy) * (wzv);                                                  \
    a0 = fmaf(w, g.x, a0);                                                    \
    a1 = fmaf(w, g.y, a1);                                                    \
  }
  CORNER(hxyA, wxyA, hz0, wz0) CORNER(hxyA, wxyA, hz1, wz)
  CORNER(hxyB, wxyB, hz0, wz0) CORNER(hxyB, wxyB, hz1, wz)
  CORNER(hxyC, wxyC, hz0, wz0) CORNER(hxyC, wxyC, hz1, wz)
  CORNER(hxyD, wxyD, hz0, wz0) CORNER(hxyD, wxyD, hz1, wz)
#undef CORNER

  out[n] = make_float2(a0, a1);
}

// ---------------------------------------------------------------------------
// Fallback: gather straight from the SoA grid (if workspace is too small).
// ---------------------------------------------------------------------------
__global__ void __launch_bounds__(BLK)
hashenc_plain(const float* __restrict__ coords,
              const float* __restrict__ grid,
              float2* __restrict__ out, int n_points) {
  __shared__ float sc[3 * BLK];
  int base = blockIdx.x * BLK;
  int cnt = n_points - base; cnt = cnt > BLK ? BLK : cnt;
  stage_coords(coords + (size_t)base * 3, sc, cnt * 3);

  int n = base + (int)threadIdx.x;
  if (n >= n_points) return;

  float cx = sc[threadIdx.x * 3 + 0];
  float cy = sc[threadIdx.x * 3 + 1];
  float cz = sc[threadIdx.x * 3 + 2];
  DECODE_POINT(cx, cy, cz);

  float a0 = 0.0f, a1 = 0.0f;
#define CORNER(hxy, wxy, hzv, wzv)                                            \
  {                                                                           \
    uint32_t h = ((hxy) ^ (hzv)) & TABLE_MASK;                                \
    float g0 = grid[h];                                                       \
    float g1 = grid[h + TABLE_SIZE];                                          \
    float w = (wxy) * (wzv);                                                  \
    a0 = fmaf(w, g0, a0);                                                     \
    a1 = fmaf(w, g1, a1);                                                     \
  }
  CORNER(hxyA, wxyA, hz0, wz0) CORNER(hxyA, wxyA, hz1, wz)
  CORNER(hxyB, wxyB, hz0, wz0) CORNER(hxyB, wxyB, hz1, wz)
  CORNER(hxyC, wxyC, hz0, wz0) CORNER(hxyC, wxyC, hz1, wz)
  CORNER(hxyD, wxyD, hz0, wz0) CORNER(hxyD, wxyD, hz1, wz)
#undef CORNER

  out[n] = make_float2(a0, a1);
}

// ---------------------------------------------------------------------------
// Host-side launch.
// ---------------------------------------------------------------------------
extern "C" void kernel_launch(void* const* d_in, const int* in_sizes, int n_in,
                              void* d_out, int out_size, void* d_ws, size_t ws_size,
                              hipStream_t stream) {
  const float* coords = (const float*)d_in[0];
  const float* grid   = (const float*)d_in[1];
  float2* out         = (float2*)d_out;

  int n_points = in_sizes[0] / 3;
  int n_blocks = (n_points + BLK - 1) / BLK;

  if (d_ws && ws_size >= (size_t)TABLE_SIZE * sizeof(float2)) {
    float2* tab = (float2*)d_ws;
    interleave_kernel<<<(TABLE_SIZE + BLK - 1) / BLK, BLK, 0, stream>>>(grid, tab);
    hashenc_interleaved<<<n_blocks, BLK, 0, stream>>>(coords, tab, out, n_points);
  } else {
    hashenc_plain<<<n_blocks, BLK, 0, stream>>>(coords, grid, out, n_points);
  }
}